// ETF_44143673868790
// MI455X (gfx1250) — compile-verified
//
#include <hip/hip_runtime.h>

typedef float v2f __attribute__((ext_vector_type(2)));
typedef float v8f __attribute__((ext_vector_type(8)));

#define MU 5
#define ITERS 3

// Hardware tanh (V_TANH_F32 transcendental) on gfx1250 if the builtin exists;
// otherwise fall back to OCML tanhf.
#if defined(__has_builtin)
#if __has_builtin(__builtin_amdgcn_tanhf)
#define FAST_TANH(x) __builtin_amdgcn_tanhf(x)
#elif __has_builtin(__builtin_amdgcn_tanh_f32)
#define FAST_TANH(x) __builtin_amdgcn_tanh_f32(x)
#endif
#endif
#ifndef FAST_TANH
#define FAST_TANH(x) tanhf(x)
#endif

// ---------------------------------------------------------------------------
// Kernel 1: Sobel -> initial tangent field + raw gradient magnitude + global max
// ---------------------------------------------------------------------------
__global__ void sobel_tang_kernel(const float* __restrict__ x,
                                  float* __restrict__ tang,   // (B,2,H,W)
                                  float* __restrict__ mag,    // (B,H,W) raw
                                  unsigned* __restrict__ gmax,
                                  int B, int H, int W) {
  const long HW = (long)H * W;
  const long total = (long)B * HW;
  long i = (long)blockIdx.x * blockDim.x + threadIdx.x;
  float m = 0.0f;
  if (i < total) {
    int b = (int)(i / HW);
    long r = i - (long)b * HW;
    int y  = (int)(r / W);
    int xx = (int)(r - (long)y * W);
    const float* img = x + (long)b * HW;
    auto at = [&](int yy, int xc) -> float {
      return (yy < 0 || yy >= H || xc < 0 || xc >= W) ? 0.0f : img[(long)yy * W + xc];
    };
    float tl = at(y - 1, xx - 1), tc = at(y - 1, xx), tr = at(y - 1, xx + 1);
    float ml = at(y,     xx - 1),                     mr = at(y,     xx + 1);
    float bl = at(y + 1, xx - 1), bc = at(y + 1, xx), br = at(y + 1, xx + 1);
    // channel 0 of sobel = k  (d/dy), channel 1 = k^T (d/dx); XLA conv = correlation
    float gy = (bl + 2.0f * bc + br) - (tl + 2.0f * tc + tr);
    float gx = (tr + 2.0f * mr + br) - (tl + 2.0f * ml + bl);
    m = sqrtf(gx * gx + gy * gy);
    float inv = (m == 0.0f) ? 1.0f : 1.0f / m;   // _normalize: mag==0 -> divide by 1
    mag[i] = m;
    float* tb = tang + (long)b * 2 * HW;
    tb[r]      = -gx * inv;   // tang component 0 = -sobel_x
    tb[HW + r] =  gy * inv;   // tang component 1 =  sobel_y
  }
  // wave-level max reduce, one atomic per wave (all values >= 0 -> uint-bit order ok)
  for (int off = 16; off > 0; off >>= 1)
    m = fmaxf(m, __shfl_xor(m, off, 32));
  if ((threadIdx.x & 31) == 0)
    atomicMax(gmax, __float_as_uint(m));
}

// ---------------------------------------------------------------------------
// Kernel 2: mag /= max(mag)
// ---------------------------------------------------------------------------
__global__ void mag_scale_kernel(float* __restrict__ mag,
                                 const unsigned* __restrict__ gmax, long n) {
  long i = (long)blockIdx.x * blockDim.x + threadIdx.x;
  if (i < n) {
    float mx = __uint_as_float(*gmax);
    float inv = (mx > 0.0f) ? (1.0f / mx) : 1.0f;
    mag[i] *= inv;
  }
}

// ---------------------------------------------------------------------------
// Kernel 3: one ETF smoothing pass (banded attention along one axis) via WMMA.
// Each wave owns a strip of 16 pixels along the pass axis at one fixed
// perpendicular coordinate. Two V_WMMA_F32_16X16X4_F32 ops compute the
// S^T = T_Q * T_P^T tiles (K=2 real components, padded to 4 with zeros in
// lanes 16..31); the data-dependent weighting + accumulation runs as a
// branch-free in-lane FMA chain using ds_bpermute broadcasts and the
// hardware V_TANH_F32 transcendental.
// ---------------------------------------------------------------------------
template <bool VERTICAL>
__global__ void etf_pass_kernel(const float* __restrict__ tin,
                                const float* __restrict__ mag,
                                float* __restrict__ tout,
                                int B, int H, int W) {
  const long HW = (long)H * W;
  const int P = VERTICAL ? H : W;   // length along pass direction
  const int Q = VERTICAL ? W : H;   // perpendicular extent
  const int stripsPerLine = P >> 4; // P / 16

  const int lane = threadIdx.x & 31;
  const int waveInBlock = threadIdx.x >> 5;
  const int wavesPerBlock = blockDim.x >> 5;
  int strip = blockIdx.x * wavesPerBlock + waveInBlock;   // exact grid, no tail

  int pstrip = strip % stripsPerLine;
  int t2 = strip / stripsPerLine;
  int qcoord = t2 % Q;
  int b = t2 / Q;

  const int p0  = pstrip << 4;
  const int l15 = lane & 15;
  const int hi  = lane >> 4;
  const bool lo = (lane < 16);

  const float* tx = tin + (long)b * 2 * HW;
  const float* ty = tx + HW;
  const float* mg = mag + (long)b * HW;

  auto pixoff = [&](int p) -> long {
    return VERTICAL ? ((long)p * W + qcoord) : ((long)qcoord * W + p);
  };

  // B operand: T_P rows p0..p0+15 (K=0 -> x comp in v0 lanes 0-15, K=1 -> y in v1)
  float bx = lo ? tx[pixoff(p0 + l15)] : 0.0f;
  float by = lo ? ty[pixoff(p0 + l15)] : 0.0f;

  // A operands: q-chunk0 = [p0-8, p0+8), q-chunk1 = [p0+8, p0+24)
  int q0 = p0 - 8 + l15;
  int q1 = p0 + 8 + l15;
  bool v0 = lo && (q0 >= 0);   // q0 <= p0+7 < P always
  bool v1 = lo && (q1 < P);    // q1 >= p0+8 >= 0 always
  float a0x = v0 ? tx[pixoff(q0)] : 0.0f;
  float a0y = v0 ? ty[pixoff(q0)] : 0.0f;
  float m0  = v0 ? mg[pixoff(q0)] : 0.0f;
  float a1x = v1 ? tx[pixoff(q1)] : 0.0f;
  float a1y = v1 ? ty[pixoff(q1)] : 0.0f;
  float m1  = v1 ? mg[pixoff(q1)] : 0.0f;

  float mp = mg[pixoff(p0 + l15)];   // mag_X, valid for all lanes

  v2f a0 = {a0x, a0y};
  v2f a1 = {a1x, a1y};
  v2f bb = {bx, by};
  v8f z  = {0.0f, 0.0f, 0.0f, 0.0f, 0.0f, 0.0f, 0.0f, 0.0f};

  // S^T tiles: D[q_rel, p_rel] = t_q . t_p   (C layout: q in vgpr j + 8*hi, p in lane&15)
  v8f S0 = __builtin_amdgcn_wmma_f32_16x16x4_f32(false, a0, false, bb, (short)0, z, false, false);
  v8f S1 = __builtin_amdgcn_wmma_f32_16x16x4_f32(false, a1, false, bb, (short)0, z, false, false);

  float accx = 0.0f, accy = 0.0f;
#pragma unroll
  for (int j = 0; j < 8; ++j) {
    int idx  = j + (hi << 3);   // source q row for this lane-half
    int bidx = idx << 2;        // byte index for ds_bpermute

    // ---- chunk 0: q = p0 - 8 + idx, delta = q - p
    {
      float txq = __int_as_float(__builtin_amdgcn_ds_bpermute(bidx, __float_as_int(a0x)));
      float tyq = __int_as_float(__builtin_amdgcn_ds_bpermute(bidx, __float_as_int(a0y)));
      float mq  = __int_as_float(__builtin_amdgcn_ds_bpermute(bidx, __float_as_int(m0)));
      int delta = -8 + idx - l15;
      bool band = (delta >= -MU) && (delta <= MU);
      // branch-free: hardware tanh is unconditionally cheap, select just the scale
      float coef = (band ? 0.5f : 0.0f) * (FAST_TANH(mq - mp) + 1.0f);
      float w = coef * S0[j];
      accx = fmaf(w, txq, accx);
      accy = fmaf(w, tyq, accy);
    }
    // ---- chunk 1: q = p0 + 8 + idx
    {
      float txq = __int_as_float(__builtin_amdgcn_ds_bpermute(bidx, __float_as_int(a1x)));
      float tyq = __int_as_float(__builtin_amdgcn_ds_bpermute(bidx, __float_as_int(a1y)));
      float mq  = __int_as_float(__builtin_amdgcn_ds_bpermute(bidx, __float_as_int(m1)));
      int delta = 8 + idx - l15;
      bool band = (delta >= -MU) && (delta <= MU);
      float coef = (band ? 0.5f : 0.0f) * (FAST_TANH(mq - mp) + 1.0f);
      float w = coef * S1[j];
      accx = fmaf(w, txq, accx);
      accy = fmaf(w, tyq, accy);
    }
  }

  // combine lane-halves (hi=0 holds q rows 0..7 of each chunk, hi=1 rows 8..15)
  accx += __shfl_xor(accx, 16, 32);
  accy += __shfl_xor(accy, 16, 32);

  float n = sqrtf(accx * accx + accy * accy);
  float inv = (n == 0.0f) ? 1.0f : 1.0f / n;

  if (lo) {
    long o = pixoff(p0 + l15);
    float* ox = tout + (long)b * 2 * HW;
    ox[o]      = accx * inv;
    ox[HW + o] = accy * inv;
  }
}

// ---------------------------------------------------------------------------
extern "C" void kernel_launch(void* const* d_in, const int* in_sizes, int n_in,
                              void* d_out, int out_size, void* d_ws, size_t ws_size,
                              hipStream_t stream) {
  const int B = 2, H = 1024, W = 1024;
  const long HW = (long)H * W;
  const float* x = (const float*)d_in[0];
  float* out = (float*)d_out;   // (B,2,H,W)

  char* ws = (char*)d_ws;
  unsigned* d_max = (unsigned*)ws;                              // 4 B (+pad)
  float* magr  = (float*)(ws + 256);                            // B*H*W f32 (8 MB)
  float* tangB = (float*)(ws + 256 + sizeof(float) * B * HW);   // B*2*H*W f32 (16 MB)
  float* tangA = out;

  hipMemsetAsync(d_max, 0, sizeof(unsigned), stream);

  {
    long total = (long)B * HW;
    int nb = (int)((total + 255) / 256);
    sobel_tang_kernel<<<nb, 256, 0, stream>>>(x, tangA, magr, d_max, B, H, W);
    mag_scale_kernel<<<nb, 256, 0, stream>>>(magr, d_max, total);
  }

  // pass kernel grid: one wave per 16-pixel strip; 4 waves / 128-thread block
  const int wavesPerBlock = 4;
  const int totalStrips = B * 1024 * (1024 / 16);   // 131072
  const int nbPass = totalStrips / wavesPerBlock;   // exact

  for (int it = 0; it < ITERS; ++it) {
    etf_pass_kernel<true ><<<nbPass, 32 * wavesPerBlock, 0, stream>>>(tangA, magr, tangB, B, H, W);
    etf_pass_kernel<false><<<nbPass, 32 * wavesPerBlock, 0, stream>>>(tangB, magr, tangA, B, H, W);
  }
  // after 3 iterations (6 passes) result is in tangA == d_out
}